// ObjectDetector_3882650435977
// MI455X (gfx1250) — compile-verified
//
#include <hip/hip_runtime.h>
#include <stdint.h>

#define B_IMG      8
#define N_ANCH     27380
#define PRE_TOPN   12000
#define POST_TOPN  2000
#define NMS_THRESH 0.7f
#define NEGV       (-1e30f)
#define NBINS      4096
#define NTHREADS   1024
#define NELEM      12          // ceil(PRE_TOPN / NTHREADS)

// ---- LDS layout (bytes); total must stay under 320KB/WGP ----
#define OFF_BOXES  0
#define OFF_SCORES (PRE_TOPN * 16)                   // 192000
#define OFF_ROIS   (OFF_SCORES + PRE_TOPN * 4)       // 240000
#define OFF_HIST   (OFF_ROIS + POST_TOPN * 5 * 4)    // 280000
#define OFF_WSUM   (OFF_HIST + NBINS * 4)            // 296384
#define OFF_RVAL   (OFF_WSUM + 128)                  // 2 x 32 floats (double buffer)
#define OFF_RIDX   (OFF_RVAL + 256)                  // 2 x 32 ints
#define OFF_SCAL   (OFF_RIDX + 256)
#define SMEM_BYTES (OFF_SCAL + 64)                   // 297088 < 327680

typedef unsigned int u32x4 __attribute__((ext_vector_type(4)));
typedef int          i32x4 __attribute__((ext_vector_type(4)));
typedef int          i32x8 __attribute__((ext_vector_type(8)));

// ---------------------------------------------------------------------------
// Kernel 1: decode boxes (streaming map)
// ---------------------------------------------------------------------------
__global__ __launch_bounds__(256)
void decode_kernel(const float* __restrict__ anchors,
                   const float* __restrict__ deltas,
                   float* __restrict__ boxes, int total) {
    int idx = blockIdx.x * blockDim.x + threadIdx.x;
    if (idx >= total) return;
    int n = idx % N_ANCH;
    const float4 a = ((const float4*)anchors)[n];
    const float4 d = ((const float4*)deltas)[idx];
    float w  = a.z - a.x + 1.0f;
    float h  = a.w - a.y + 1.0f;
    float cx = a.x + 0.5f * w;
    float cy = a.y + 0.5f * h;
    float px = cx + w * d.x;
    float py = cy + h * d.y;
    float pw = __expf(d.z) * w;
    float ph = __expf(d.w) * h;
    float4 o;
    o.x = px - 0.5f * pw;
    o.y = py - 0.5f * ph;
    o.z = px + 0.5f * (pw - 2.0f);
    o.w = py + 0.5f * (ph - 2.0f);
    ((float4*)boxes)[idx] = o;
}

// ---------------------------------------------------------------------------
// Block-wide inclusive scan (wave32 shfl scan + wave-total scan).
// Leaves s_wsum reusable (trailing barrier). Returns inclusive prefix; total
// of whole block via `total`.
// ---------------------------------------------------------------------------
__device__ __forceinline__ int block_scan_incl(int v, int lane, int wave,
                                               int* s_wsum, int& total) {
    for (int d = 1; d < 32; d <<= 1) {
        int nv = __shfl_up(v, d);
        if (lane >= d) v += nv;
    }
    if (lane == 31) s_wsum[wave] = v;
    __syncthreads();
    if (wave == 0) {
        int wv = s_wsum[lane];
        for (int d = 1; d < 32; d <<= 1) {
            int nv = __shfl_up(wv, d);
            if (lane >= d) wv += nv;
        }
        s_wsum[lane] = wv;
    }
    __syncthreads();
    int incl = v + (wave ? s_wsum[wave - 1] : 0);
    total = s_wsum[31];
    __syncthreads();
    return incl;
}

// ---------------------------------------------------------------------------
// Kernel 2: per-image top-K + greedy NMS; candidate set register-resident,
// boxes staged in LDS via CDNA5 async gather, result via TDM store.
// ---------------------------------------------------------------------------
__global__ __launch_bounds__(NTHREADS)
void rpn_nms_kernel(const float* __restrict__ boxes_all,   // [B,N,4]
                    const float* __restrict__ scores_all,  // [B,N]
                    float* __restrict__ out)               // [B*POST,5]
{
    extern __shared__ char smem[];
    float4* s_box  = (float4*)(smem + OFF_BOXES);
    float*  s_sc   = (float*) (smem + OFF_SCORES);
    float*  s_roi  = (float*) (smem + OFF_ROIS);
    int*    s_hist = (int*)   (smem + OFF_HIST);
    int*    s_wsum = (int*)   (smem + OFF_WSUM);
    float*  s_rval = (float*) (smem + OFF_RVAL);
    int*    s_ridx = (int*)   (smem + OFF_RIDX);
    int*    s_scal = (int*)   (smem + OFF_SCAL);

    const int b    = blockIdx.x;
    const int tid  = threadIdx.x;
    const int lane = tid & 31;
    const int wave = tid >> 5;
    const float* sc_g = scores_all + (size_t)b * N_ANCH;
    const float* bx_g = boxes_all  + (size_t)b * N_ANCH * 4;

    // ---- Phase 0/1: score histogram ----
    for (int i = tid; i < NBINS; i += NTHREADS) s_hist[i] = 0;
    __syncthreads();
    for (int i = tid; i < N_ANCH; i += NTHREADS) {
        float s = sc_g[i];
        int bin = (int)(s * (float)NBINS);
        bin = bin < 0 ? 0 : (bin > NBINS - 1 ? NBINS - 1 : bin);
        atomicAdd(&s_hist[bin], 1);
    }
    __syncthreads();

    // ---- Phase 2: parallel threshold search (suffix sums, reversed bins) ----
    {
        int h[4];
        int lsum = 0;
        #pragma unroll
        for (int s = 0; s < 4; ++s) {
            int bn = NBINS - 1 - (tid * 4 + s);
            h[s] = s_hist[bn];
            lsum += h[s];
        }
        int total;
        int incl = block_scan_incl(lsum, lane, wave, s_wsum, total);
        int excl = incl - lsum;
        if (excl < PRE_TOPN && incl >= PRE_TOPN) {   // unique crossing thread
            int cum = excl;
            #pragma unroll
            for (int s = 0; s < 4; ++s) {
                int bn = NBINS - 1 - (tid * 4 + s);
                if (cum + h[s] >= PRE_TOPN) { s_scal[0] = bn; s_scal[1] = cum; break; }
                cum += h[s];
            }
        }
        if (tid == 0) s_scal[2] = 0;
        __syncthreads();
    }
    const int tbin   = s_scal[0];
    const int nAbove = s_scal[1];

    // ---- Phase 3: deterministic chunked compaction (original index order) ----
    for (int base = 0; base < N_ANCH; base += NTHREADS) {
        int i = base + tid;
        int predA = 0, predB = 0;
        float s = 0.0f;
        if (i < N_ANCH) {
            s = sc_g[i];
            int bin = (int)(s * (float)NBINS);
            bin = bin < 0 ? 0 : (bin > NBINS - 1 ? NBINS - 1 : bin);
            predA = (bin > tbin) ? 1 : 0;
            predB = (bin == tbin) ? 1 : 0;
        }
        int p = predA | (predB << 16);
        int total;
        int incl = block_scan_incl(p, lane, wave, s_wsum, total);
        int excl = incl - p;
        int run  = s_scal[2];

        int pos = -1;
        if (predA) {
            pos = (run & 0xffff) + (excl & 0xffff);
        } else if (predB) {
            pos = nAbove + (run >> 16) + (excl >> 16);
            if (pos >= PRE_TOPN) pos = -1;
        }
        if (pos >= 0) {
            s_sc[pos] = s;
            // CDNA5 async gather: 16B box straight from HBM/L2 into its LDS slot
            unsigned lds_off   = (unsigned)(uintptr_t)(&s_box[pos]);
            const float* gaddr = bx_g + (size_t)i * 4;
            asm volatile("global_load_async_to_lds_b128 %0, %1, off"
                         :: "v"(lds_off), "v"(gaddr) : "memory");
        }
        __syncthreads();
        if (tid == 0) s_scal[2] = run + total;   // packed fields cannot cross-carry
    }

    // drain async copies, make LDS boxes visible to all waves
    asm volatile("s_wait_asynccnt 0x0" ::: "memory");
    __syncthreads();

    // ---- Phase 4: pull candidate set into registers (12 elems/thread) ----
    float  rs[NELEM];
    float4 rb[NELEM];
    float  ra[NELEM];
    #pragma unroll
    for (int m = 0; m < NELEM; ++m) {
        int j = tid + m * NTHREADS;
        if (j < PRE_TOPN) {
            rs[m] = s_sc[j];
            rb[m] = s_box[j];
        } else {
            rs[m] = NEGV;
            rb[m] = make_float4(0.0f, 0.0f, 0.0f, 0.0f);
        }
        ra[m] = (rb[m].z - rb[m].x + 1.0f) * (rb[m].w - rb[m].y + 1.0f);
    }

    // ---- Phase 5: greedy NMS, 2000 picks, 2 barriers/iteration ----
    for (int k = 0; k < POST_TOPN; ++k) {
        float bv = NEGV;
        int   bi = 0x7fffffff;
        #pragma unroll
        for (int m = 0; m < NELEM; ++m) {
            int j = tid + m * NTHREADS;
            float sv = rs[m];
            if (sv > bv || (sv == bv && j < bi)) { bv = sv; bi = j; }
        }
        for (int d = 16; d > 0; d >>= 1) {
            float ov = __shfl_down(bv, d);
            int   oi = __shfl_down(bi, d);
            if (ov > bv || (ov == bv && oi < bi)) { bv = ov; bi = oi; }
        }
        float* rv = s_rval + (k & 1) * 32;       // double-buffered slots
        int*   ri = s_ridx + (k & 1) * 32;
        if (lane == 0) { rv[wave] = bv; ri[wave] = bi; }
        __syncthreads();
        if (wave == 0) {
            bv = rv[lane];
            bi = ri[lane];
            for (int d = 16; d > 0; d >>= 1) {
                float ov = __shfl_down(bv, d);
                int   oi = __shfl_down(bi, d);
                if (ov > bv || (ov == bv && oi < bi)) { bv = ov; bi = oi; }
            }
            if (lane == 0) { rv[0] = bv; ri[0] = bi; }
        }
        __syncthreads();

        const int   isel = ri[0];
        const float vsel = rv[0];
        if (!(vsel > (NEGV * 0.5f))) {
            // candidates exhausted: zero-fill remaining ROI slots and stop
            for (int q = k * 5 + tid; q < POST_TOPN * 5; q += NTHREADS)
                s_roi[q] = 0.0f;
            break;
        }
        const float4 bb = s_box[isel];           // LDS broadcast (single bank)
        if (tid == 0) {
            float* r = s_roi + k * 5;
            r[0] = (float)b; r[1] = bb.x; r[2] = bb.y; r[3] = bb.z; r[4] = bb.w;
        }
        const float areaI = (bb.z - bb.x + 1.0f) * (bb.w - bb.y + 1.0f);
        #pragma unroll
        for (int m = 0; m < NELEM; ++m) {
            float4 bj = rb[m];
            float xx1 = fmaxf(bb.x, bj.x);
            float yy1 = fmaxf(bb.y, bj.y);
            float xx2 = fminf(bb.z, bj.z);
            float yy2 = fminf(bb.w, bj.w);
            float iw  = fmaxf(xx2 - xx1 + 1.0f, 0.0f);
            float ih  = fmaxf(yy2 - yy1 + 1.0f, 0.0f);
            float inter = iw * ih;
            // iou > t  <=>  inter > t * union   (union > 0 always)
            if (inter > NMS_THRESH * (areaI + ra[m] - inter)) rs[m] = NEGV;
        }
    }
    __syncthreads();

    // ---- Phase 6: bulk ROI writeback via Tensor Data Mover (one D#) ----
    if (wave == 0) {
        const unsigned nelem = POST_TOPN * 5;                 // 10000 f32
        unsigned lds_addr = (unsigned)(uintptr_t)s_roi;
        unsigned long long ga =
            (unsigned long long)(uintptr_t)(out + (size_t)b * nelem);
        u32x4 g0;
        g0.x = 1u;                                               // count=1 (valid D#)
        g0.y = lds_addr;                                         // lds_addr
        g0.z = (unsigned)(ga & 0xffffffffu);                     // global_addr[31:0]
        g0.w = (unsigned)((ga >> 32) & 0x1ffffffu) | (2u << 30); // addr[56:32]+type=2
        i32x8 g1 = {0, 0, 0, 0, 0, 0, 0, 0};
        g1[0] = (int)(2u << 16);                                 // data_size = 4B
        g1[1] = (int)((nelem & 0xffffu) << 16);                  // tensor_dim0 lo16
        g1[2] = (int)(((nelem >> 16) & 0xffffu) | (1u << 16));   // dim0 hi + tensor_dim1=1
        g1[3] = (int)(nelem << 16);                              // tile_dim0 = 10000
        g1[4] = 1;                                               // tile_dim1 = 1
        g1[5] = (int)nelem;                                      // tensor_dim0_stride
        i32x4 gz4 = {0, 0, 0, 0};
#if defined(__clang_major__) && (__clang_major__ >= 23)
        i32x8 gz8 = {0, 0, 0, 0, 0, 0, 0, 0};
        __builtin_amdgcn_tensor_store_from_lds(g0, g1, gz4, gz4, gz8, 0);
#else
        __builtin_amdgcn_tensor_store_from_lds(g0, g1, gz4, gz4, 0);
#endif
        __builtin_amdgcn_s_wait_tensorcnt(0);
    }
}

// ---------------------------------------------------------------------------
extern "C" void kernel_launch(void* const* d_in, const int* in_sizes, int n_in,
                              void* d_out, int out_size, void* d_ws, size_t ws_size,
                              hipStream_t stream) {
    const float* anchors = (const float*)d_in[0];   // [N,4]
    const float* deltas  = (const float*)d_in[1];   // [B,N,4]
    const float* scores  = (const float*)d_in[2];   // [B,N]
    float* out   = (float*)d_out;                   // [B*POST,5]
    float* boxes = (float*)d_ws;                    // [B,N,4] scratch (3.5 MB)

    const int total = B_IMG * N_ANCH;
    decode_kernel<<<(total + 255) / 256, 256, 0, stream>>>(anchors, deltas, boxes, total);

    static_assert(SMEM_BYTES <= 320 * 1024, "LDS budget exceeded");
    hipFuncSetAttribute(reinterpret_cast<const void*>(rpn_nms_kernel),
                        hipFuncAttributeMaxDynamicSharedMemorySize, SMEM_BYTES);
    rpn_nms_kernel<<<B_IMG, NTHREADS, SMEM_BYTES, stream>>>(boxes, scores, out);
}